// SpatialTransformer_87582973100656
// MI455X (gfx1250) — compile-verified
//
#include <hip/hip_runtime.h>
#include <hip/hip_bf16.h>

// SpatialTransformer forward: identity grid + flow, bilinear sample with
// zero padding (align_corners=True). Memory-bound gather: ~264 MB traffic,
// ~11.3 us floor at 23.3 TB/s HBM. fp32 throughout (reference dtype).

#define ST_N 4
#define ST_C 32
#define ST_H 512
#define ST_W 512
#define ST_HW (ST_H * ST_W)

// Address-space-qualified pointer types for the gfx1250 async-to-LDS builtin.
// Compiler diagnostic showed param 0 is an AS(1) (global) non-const int*.
typedef __attribute__((address_space(1))) int as1_int;
typedef __attribute__((address_space(3))) int as3_int;

__global__ __launch_bounds__(256) void spatial_transformer_kernel(
    const float* __restrict__ src,   // (N, C, H, W)
    const float* __restrict__ flow,  // (N, 2, H, W)
    float* __restrict__ out)         // (N, C, H, W)
{
    const int tid = threadIdx.x;
    const int pix = blockIdx.x * 256 + tid;       // linear (n, h, w)
    const int w = pix & (ST_W - 1);
    const int h = (pix >> 9) & (ST_H - 1);
    const int n = pix >> 18;

    const int fbase = n * 2 * ST_HW + h * ST_W + w;

    float fx, fy;
#if __has_builtin(__builtin_amdgcn_global_load_async_to_lds_b32)
    // Stage flow through LDS via the gfx1250 async-to-LDS path (ASYNCcnt).
    __shared__ float sfx[256];
    __shared__ float sfy[256];
    __builtin_amdgcn_global_load_async_to_lds_b32(
        (as1_int*)(flow + fbase),
        (as3_int*)&sfx[tid], 0, 0);
    __builtin_amdgcn_global_load_async_to_lds_b32(
        (as1_int*)(flow + fbase + ST_HW),
        (as3_int*)&sfy[tid], 0, 0);
  #if __has_builtin(__builtin_amdgcn_s_wait_asynccnt)
    __builtin_amdgcn_s_wait_asynccnt(0);
  #else
    asm volatile("s_wait_asynccnt 0" ::: "memory");
  #endif
    fx = sfx[tid];
    fy = sfy[tid];
#else
    fx = flow[fbase];
    fy = flow[fbase + ST_HW];
#endif

    // Replicate the reference float arithmetic exactly:
    // normalize to [-1,1] then unnormalize (align_corners=True round trip).
    const float newx = (float)w + fx;
    const float newy = (float)h + fy;
    const float xn = 2.0f * (newx / (float)(ST_W - 1) - 0.5f);
    const float yn = 2.0f * (newy / (float)(ST_H - 1) - 0.5f);
    const float x = (xn + 1.0f) / 2.0f * (float)(ST_W - 1);
    const float y = (yn + 1.0f) / 2.0f * (float)(ST_H - 1);

    const float x0f = floorf(x);
    const float y0f = floorf(y);
    const float x1f = x0f + 1.0f;
    const float y1f = y0f + 1.0f;

    const float wx0 = x1f - x;   // weight toward x0
    const float wx1 = x - x0f;   // weight toward x1
    const float wy0 = y1f - y;
    const float wy1 = y - y0f;

    float wa = wx0 * wy0;   // (x0, y0)
    float wb = wx0 * wy1;   // (x0, y1)
    float wc = wx1 * wy0;   // (x1, y0)
    float wd = wx1 * wy1;   // (x1, y1)

    const int x0 = (int)x0f;
    const int y0 = (int)y0f;
    const int x1 = x0 + 1;
    const int y1 = y0 + 1;

    // Zero-pad semantics: any tap outside [0,W)x[0,H) contributes 0.
    // Fold the mask into the weights (branchless), clamp indices for safety.
    const bool vx0 = (unsigned)x0 < (unsigned)ST_W;
    const bool vx1 = (unsigned)x1 < (unsigned)ST_W;
    const bool vy0 = (unsigned)y0 < (unsigned)ST_H;
    const bool vy1 = (unsigned)y1 < (unsigned)ST_H;
    wa = (vx0 && vy0) ? wa : 0.0f;
    wb = (vx0 && vy1) ? wb : 0.0f;
    wc = (vx1 && vy0) ? wc : 0.0f;
    wd = (vx1 && vy1) ? wd : 0.0f;

    const int xc0 = min(max(x0, 0), ST_W - 1);
    const int xc1 = min(max(x1, 0), ST_W - 1);
    const int yc0 = min(max(y0, 0), ST_H - 1);
    const int yc1 = min(max(y1, 0), ST_H - 1);

    const size_t base = (size_t)n * ST_C * ST_HW;
    const float* pA = src + base + (size_t)yc0 * ST_W + xc0;
    const float* pB = src + base + (size_t)yc1 * ST_W + xc0;
    const float* pC = src + base + (size_t)yc0 * ST_W + xc1;
    const float* pD = src + base + (size_t)yc1 * ST_W + xc1;
    float* pO = out + base + (size_t)h * ST_W + w;

    // Per-pixel weights reused across all 32 channels (stride H*W = 1 MB).
#pragma unroll 8
    for (int c = 0; c < ST_C; ++c) {
        // Pull the next channel group's tap cachelines toward the WGP while
        // this group's loads are in flight (gfx1250 global_prefetch_b8).
        if ((c & 7) == 0 && (c + 8) < ST_C) {
            __builtin_prefetch(pA + (size_t)(c + 8) * ST_HW, 0, 0);
            __builtin_prefetch(pB + (size_t)(c + 8) * ST_HW, 0, 0);
            __builtin_prefetch(pC + (size_t)(c + 8) * ST_HW, 0, 0);
            __builtin_prefetch(pD + (size_t)(c + 8) * ST_HW, 0, 0);
        }
        const size_t o = (size_t)c * ST_HW;
        const float va = pA[o];
        const float vb = pB[o];
        const float vc = pC[o];
        const float vd = pD[o];
        const float r = va * wa + vb * wb + vc * wc + vd * wd;
        // Output is write-once streamed: non-temporal so it doesn't evict
        // reusable src lines from the 192 MB L2.
        __builtin_nontemporal_store(r, pO + o);
    }
}

extern "C" void kernel_launch(void* const* d_in, const int* in_sizes, int n_in,
                              void* d_out, int out_size, void* d_ws, size_t ws_size,
                              hipStream_t stream) {
    const float* src = (const float*)d_in[0];   // (4, 32, 512, 512) fp32
    const float* flow = (const float*)d_in[1];  // (4, 2, 512, 512) fp32
    float* out = (float*)d_out;                 // (4, 32, 512, 512) fp32

    const int pixels = ST_N * ST_H * ST_W;      // 1,048,576
    dim3 block(256);                            // 8 wave32 waves per block
    dim3 grid(pixels / 256);                    // 4096 blocks
    spatial_transformer_kernel<<<grid, block, 0, stream>>>(src, flow, out);
}